// MemoryModel_11793980195326
// MI455X (gfx1250) — compile-verified
//
#include <hip/hip_runtime.h>
#include <hip/hip_bf16.h>
#include <stdint.h>

// ---------------------------------------------------------------------------
// Discounted cumulative return (reverse linear-recurrence scan), MI455X.
//   y[i] = r[i] + a[i]*y[i+1],  a[i] = terminal[i] ? 0 : 0.99,  y[T] = 0
// 3-pass affine scan; tiles staged to LDS via the Tensor Data Mover with
// the D# pad feature producing a bank-conflict-free skewed layout.
// ---------------------------------------------------------------------------

#define TILE   4096   // elements per block
#define BLOCK  256    // threads per block (8 waves of 32)
#define CHUNK  16     // elements per thread
#define GAMMA  0.99f

// Skewed LDS layouts (what TDM padding produces):
//   reward  (4B elems): dword index l -> l + (l>>4)        (pad 1 dword / 16 dwords)
//   terminal(1B elems): byte  index l -> l + ((l>>4)<<2)   (pad 4 bytes / 16 bytes)
#define RSKEW(l) ((l) + ((l) >> 4))
#define TSKEW(l) ((l) + (((l) >> 4) << 2))

// ------------------------------- TDM support -------------------------------
#if defined(__has_builtin)
#if __has_builtin(__builtin_amdgcn_tensor_load_to_lds) && \
    __has_builtin(__builtin_amdgcn_s_wait_tensorcnt)
#define USE_TDM 1
#endif
#endif
#ifndef USE_TDM
#define USE_TDM 0
#endif

typedef unsigned int v4u __attribute__((ext_vector_type(4)));
typedef int          v8i __attribute__((ext_vector_type(8)));
typedef int          v4i __attribute__((ext_vector_type(4)));

__device__ __forceinline__ uint32_t lds_addr_of(const void* p) {
  // Generic pointers into the LDS aperture carry the LDS byte offset in the
  // low 32 bits.
  return (uint32_t)(uintptr_t)p;
}

#if USE_TDM
// 1-row tile load Global->LDS through the Tensor Data Mover.
//  n_elems       : tile/tensor length (fits tile_dim0's 16 bits)
//  data_size_code: 0=1B, 1=2B, 2=4B, 3=8B
//  pad_code      : D#.pad_interval code (0=2 dwords,1=4,...,3=16); pad_amount=1 dword
__device__ __forceinline__ void tdm_load_1d(const void* gptr, uint32_t lds_addr,
                                            uint32_t n_elems, uint32_t data_size_code,
                                            uint32_t pad_code) {
  uint64_t ga = (uint64_t)(uintptr_t)gptr;
  v4u g0;
  g0[0] = 1u;                                   // count=1, not restore, no gather
  g0[1] = lds_addr;                             // lds_addr[31:0]
  g0[2] = (uint32_t)(ga & 0xFFFFFFFFu);         // global_addr[31:0]
  g0[3] = (uint32_t)((ga >> 32) & 0x01FFFFFFu)  // global_addr[56:32]
          | (2u << 30);                         // type = 2 ("image")
  v8i g1;
  g1[0] = (int)((data_size_code << 16)          // data_size
          | (1u << 20)                          // pad_enable
          | (pad_code << 22));                  // pad_interval; pad_amount=0 (=1 dword)
  g1[1] = (int)((n_elems & 0xFFFFu) << 16);     // tensor_dim0[15:0] @ bits[63:48]
  g1[2] = (int)(((n_elems >> 16) & 0xFFFFu)     // tensor_dim0[31:16]
          | (1u << 16));                        // tensor_dim1 = 1
  g1[3] = (int)((n_elems & 0xFFFFu) << 16);     // tile_dim0 = n_elems
  g1[4] = 1;                                    // tile_dim1 = 1, tile_dim2 = 0
  g1[5] = (int)n_elems;                         // tensor_dim0_stride[31:0]
  g1[6] = 0;                                    // stride hi / dim1_stride lo
  g1[7] = 0;
  v4i z4 = {0, 0, 0, 0};
#if defined(__clang_major__) && (__clang_major__ >= 23)
  v8i z8 = {0, 0, 0, 0, 0, 0, 0, 0};
  __builtin_amdgcn_tensor_load_to_lds(g0, g1, z4, z4, z8, 0);
#else
  __builtin_amdgcn_tensor_load_to_lds(g0, g1, z4, z4, 0);
#endif
}
#endif  // USE_TDM

// Stage one tile (reward + terminal) into skewed LDS and sync the block.
__device__ __forceinline__ void stage_tiles(const float* __restrict__ rew,
                                            const unsigned char* __restrict__ term,
                                            int i_base, float* rt,
                                            unsigned char* tb, int tid) {
#if USE_TDM
  if (tid == 0) {                 // one wave issues both DMAs (EXEC ignored by TDM)
    tdm_load_1d(rew + i_base, lds_addr_of(rt), TILE, /*4B*/ 2u, /*16 dw*/ 3u);
    tdm_load_1d(term + i_base, lds_addr_of(tb), TILE, /*1B*/ 0u, /*4 dw*/ 1u);
  }
  __builtin_amdgcn_s_wait_tensorcnt(0);
#else
  for (int l = tid; l < TILE; l += BLOCK) {
    rt[RSKEW(l)] = rew[i_base + l];
    tb[TSKEW(l)] = term[i_base + l];
  }
#endif
  __syncthreads();
}

// Inclusive Hillis-Steele scan of 256 affine pairs in LDS.
// Thread order == scan order (thread 0 earliest). combine(e1,e2)=(a1*a2, a2*b1+b2).
__device__ __forceinline__ void lds_scan_inclusive(float* sA, float* sB, int t) {
#pragma unroll
  for (int off = 1; off < BLOCK; off <<= 1) {
    float a1 = 1.f, b1 = 0.f;
    const bool act = (t >= off);
    if (act) { a1 = sA[t - off]; b1 = sB[t - off]; }
    __syncthreads();
    if (act) {
      float a2 = sA[t], b2 = sB[t];
      sA[t] = a1 * a2;
      sB[t] = fmaf(a2, b1, b2);
    }
    __syncthreads();
  }
}

// Per-thread affine aggregate over its 16-element chunk, ascending in scan
// order j (descending in memory order i). LDS idx l = TILE-1 - j_local.
__device__ __forceinline__ void thread_aggregate(const float* rt,
                                                 const unsigned char* tb, int t,
                                                 float& A, float& B) {
  A = 1.f; B = 0.f;
  const int l0 = TILE - 1 - CHUNK * t;
#pragma unroll
  for (int k = 0; k < CHUNK; ++k) {
    const int l = l0 - k;
    const float b = rt[RSKEW(l)];
    const float a = tb[TSKEW(l)] ? 0.f : GAMMA;
    B = fmaf(a, B, b);
    A *= a;
  }
}

// --------------------------- Kernel 1: block aggregates ---------------------
__global__ __launch_bounds__(BLOCK) void k1_block_agg(
    const unsigned char* __restrict__ term, const float* __restrict__ rew,
    float* __restrict__ blockA, float* __restrict__ blockB, int T) {
  __shared__ float rt[TILE + TILE / 16];
  __shared__ unsigned char tb[TILE + (TILE / 16) * 4];
  __shared__ float sA[BLOCK], sB[BLOCK];
  const int p = blockIdx.x, t = threadIdx.x;
  const int i_base = T - (p + 1) * TILE;

  stage_tiles(rew, term, i_base, rt, tb, t);

  float A, B;
  thread_aggregate(rt, tb, t, A, B);
  sA[t] = A; sB[t] = B;
  __syncthreads();
  lds_scan_inclusive(sA, sB, t);
  if (t == BLOCK - 1) { blockA[p] = sA[t]; blockB[p] = sB[t]; }
}

// --------------------------- Kernel 2: scan block aggregates ----------------
// One block; nb <= BLOCK*CHUNK (= 4096) entries. carry[p] = y entering block p.
__global__ __launch_bounds__(BLOCK) void k2_scan_blocks(
    const float* __restrict__ blockA, const float* __restrict__ blockB,
    float* __restrict__ carry, int nb) {
  __shared__ float sA[BLOCK], sB[BLOCK];
  const int t = threadIdx.x, base = t * CHUNK;
  float ea[CHUNK], eb[CHUNK];
  float A = 1.f, B = 0.f;
#pragma unroll
  for (int k = 0; k < CHUNK; ++k) {
    const int idx = base + k;
    float a = 1.f, b = 0.f;                       // identity for out-of-range
    if (idx < nb) { a = blockA[idx]; b = blockB[idx]; }
    ea[k] = a; eb[k] = b;
    B = fmaf(a, B, b);
    A *= a;
  }
  sA[t] = A; sB[t] = B;
  __syncthreads();
  lds_scan_inclusive(sA, sB, t);
  float pA = 1.f, pB = 0.f;                       // exclusive thread prefix
  if (t > 0) { pA = sA[t - 1]; pB = sB[t - 1]; }
#pragma unroll
  for (int k = 0; k < CHUNK; ++k) {
    const int idx = base + k;
    if (idx < nb) carry[idx] = pB;                // seed y is 0 -> only B matters
    pB = fmaf(ea[k], pB, eb[k]);                  // prefix = combine(prefix, entry)
    pA *= ea[k];
  }
}

// --------------------------- Kernel 3: finalize -----------------------------
__global__ __launch_bounds__(BLOCK) void k3_finalize(
    const unsigned char* __restrict__ term, const float* __restrict__ rew,
    const float* __restrict__ carry, float* __restrict__ out, int T) {
  __shared__ float rt[TILE + TILE / 16];
  __shared__ unsigned char tb[TILE + (TILE / 16) * 4];
  __shared__ float sA[BLOCK], sB[BLOCK];
  const int p = blockIdx.x, t = threadIdx.x;
  const int i_base = T - (p + 1) * TILE;

  stage_tiles(rew, term, i_base, rt, tb, t);

  float A, B;
  thread_aggregate(rt, tb, t, A, B);
  sA[t] = A; sB[t] = B;
  __syncthreads();
  lds_scan_inclusive(sA, sB, t);
  float eA = 1.f, eB = 0.f;
  if (t > 0) { eA = sA[t - 1]; eB = sB[t - 1]; }
  float y = fmaf(eA, carry[p], eB);               // value entering this chunk

  // Emit y's into the skewed tile (each thread touches only its own chunk).
  const int l0 = TILE - 1 - CHUNK * t;
#pragma unroll
  for (int k = 0; k < CHUNK; ++k) {
    const int l = l0 - k;
    const float b = rt[RSKEW(l)];
    const float a = tb[TSKEW(l)] ? 0.f : GAMMA;
    y = fmaf(a, y, b);
    rt[RSKEW(l)] = y;
  }
  __syncthreads();

  // Coalesced write-out in ascending i order: 4x float4 per thread.
#pragma unroll
  for (int q = 0; q < 4; ++q) {
    const int li = CHUNK * t + 4 * q;             // (li..li+3) share one 16-group
    float4 v;
    v.x = rt[RSKEW(li + 0)];
    v.y = rt[RSKEW(li + 1)];
    v.z = rt[RSKEW(li + 2)];
    v.w = rt[RSKEW(li + 3)];
    *reinterpret_cast<float4*>(out + i_base + li) = v;
  }
}

// --------------------------- Launcher ---------------------------------------
extern "C" void kernel_launch(void* const* d_in, const int* in_sizes, int n_in,
                              void* d_out, int out_size, void* d_ws, size_t ws_size,
                              hipStream_t stream) {
  const unsigned char* term = (const unsigned char*)d_in[0];  // bool -> 1 byte
  const float* rew = (const float*)d_in[1];
  float* out = (float*)d_out;
  const int T = in_sizes[1];          // 16777216, multiple of TILE
  const int nb = T / TILE;            // 4096 blocks (== BLOCK*CHUNK for k2)

  float* blockA = (float*)d_ws;       // nb floats
  float* blockB = blockA + nb;        // nb floats
  float* carry  = blockB + nb;        // nb floats (48 KB total)

  k1_block_agg<<<nb, BLOCK, 0, stream>>>(term, rew, blockA, blockB, T);
  k2_scan_blocks<<<1, BLOCK, 0, stream>>>(blockA, blockB, carry, nb);
  k3_finalize<<<nb, BLOCK, 0, stream>>>(term, rew, carry, out, T);
}